// QuanvolutionClassifierQuantum_65481071400545
// MI455X (gfx1250) — compile-verified
//
#include <hip/hip_runtime.h>
#include <hip/hip_bf16.h>
#include <math.h>

// Quanvolution collapses analytically (disjoint-wire circuit => product state),
// leaving: per-patch cosines + a (4x1024)@(1024x10) head done with f32 WMMA.

typedef float v2f __attribute__((ext_vector_type(2)));
typedef float v8f __attribute__((ext_vector_type(8)));

#define DIMK 1024   // feature length per image (64 patches * 16 wires)

__global__ __launch_bounds__(256)
void quanv_head_kernel(const float* __restrict__ x,      // (4,4,16,16)
                       const float* __restrict__ qp,     // (5,)
                       const float* __restrict__ hw,     // (10,1024)
                       const float* __restrict__ hb,     // (10,)
                       float* __restrict__ out)          // (4,10) log-softmax
{
    __shared__ float featS[4 * DIMK];   // (4 images, 1024 features)
    __shared__ float logitS[4 * 16];    // padded logits

    const int t = threadIdx.x;

    // ---------------- Phase 1: closed-form <Z_i> per patch ----------------
    {
        const int b  = t >> 6;        // image
        const int p  = t & 63;        // patch id
        const int py = p >> 3;
        const int px = p & 7;

        float ang[16];
        #pragma unroll
        for (int cch = 0; cch < 4; ++cch)
            #pragma unroll
            for (int dy = 0; dy < 2; ++dy)
                #pragma unroll
                for (int dx = 0; dx < 2; ++dx) {
                    const int w = cch * 4 + dy * 2 + dx;
                    ang[w] = x[((b * 4 + cch) * 16 + (2 * py + dy)) * 16 + (2 * px + dx)];
                }

        const float p0 = qp[0], p1 = qp[1], p3 = qp[3], p4 = qp[4]; // qp[2] (RZ) drops out

        float m[16];
        m[0]  = cosf(ang[0]);                 // CNOT(0,8) control
        m[9]  = cosf(ang[9]);                 // CNOT(9,1) control
        m[5]  = cosf(ang[5]);                 // CNOT(5,12) control
        m[1]  = m[9] * cosf(ang[1]);          // CNOT target
        m[2]  = cosf(ang[2] + p4);            // RY(p4) after RY(x2)
        m[3]  = cosf(p0) * cosf(ang[3]);      // RX(p0) after RY(x3)
        m[4]  = cosf(ang[4]);
        m[6]  = cosf(ang[6]);
        m[7]  = cosf(ang[7] + p1);            // RY(p1) after RY(x7)
        m[8]  = m[0] * cosf(ang[8]);          // CNOT target
        m[10] = cosf(ang[10]);
        m[11] = cosf(ang[11]);                // RZ leaves <Z> = cos(x11)
        m[12] = m[5] * cosf(ang[12]);         // CNOT target
        m[13] = cosf(ang[13]);
        m[14] = cosf(p3) * cosf(ang[14]);     // RX(p3) after RY(x14)
        m[15] = cosf(ang[15]);

        #pragma unroll
        for (int w = 0; w < 16; ++w)
            featS[t * 16 + w] = m[w];         // feat[b][p*16+w] == featS[t*16+w]
    }
    __syncthreads();

    // ---------------- Phase 2: head GEMM via V_WMMA_F32_16X16X4_F32 --------
    // D(16x16) = A(16x4) x B(4x16) + C, chained over K=1024 (256 steps).
    // A rows 0..3 = feat per image (rows 4..15 zero); B cols 0..9 = head_w rows.
    if (t < 32) {
        const int lane = t;
        const int half = lane >> 4;       // 0: K={0,1}, 1: K={2,3}
        const int nm   = lane & 15;       // A row (M) / B col (N) for this lane

        const float amf = (nm < 4)  ? 1.0f : 0.0f;   // branchless zero-padding
        const int   ar  = (nm < 4)  ? nm   : 0;      // (keeps EXEC all-ones)
        const float bmf = (nm < 10) ? 1.0f : 0.0f;
        const int   br  = (nm < 10) ? nm   : 0;

        const float* fRow = &featS[ar * DIMK];   // feat row (LDS)
        const float* wRow = &hw[br * DIMK];      // head_w row (global)

        v8f acc = {0.f, 0.f, 0.f, 0.f, 0.f, 0.f, 0.f, 0.f};

        #pragma unroll 4
        for (int kb = 0; kb < DIMK; kb += 4) {
            const int k0 = kb + 2 * half;
            v2f a, bv;
            a.x  = fRow[k0]     * amf;
            a.y  = fRow[k0 + 1] * amf;
            bv.x = wRow[k0]     * bmf;
            bv.y = wRow[k0 + 1] * bmf;
            acc = __builtin_amdgcn_wmma_f32_16x16x4_f32(
                      /*neg_a=*/false, a, /*neg_b=*/false, bv,
                      /*c_mod=*/(short)0, acc,
                      /*reuse_a=*/false, /*reuse_b=*/false);
        }

        // D layout: VGPR r -> row M=r on lanes 0..15 (N=lane). Batch rows 0..3.
        if (lane < 16) {
            const float bias = (lane < 10) ? hb[lane] : 0.0f;
            logitS[0 * 16 + lane] = acc[0] + bias;
            logitS[1 * 16 + lane] = acc[1] + bias;
            logitS[2 * 16 + lane] = acc[2] + bias;
            logitS[3 * 16 + lane] = acc[3] + bias;
        }
    }
    __syncthreads();

    // ---------------- Phase 3: log-softmax over 10 classes ----------------
    if (t < 4) {
        float v[10];
        float mx = -INFINITY;
        #pragma unroll
        for (int k = 0; k < 10; ++k) {
            v[k] = logitS[t * 16 + k];
            mx = fmaxf(mx, v[k]);
        }
        float s = 0.0f;
        #pragma unroll
        for (int k = 0; k < 10; ++k)
            s += expf(v[k] - mx);
        const float lse = mx + logf(s);
        #pragma unroll
        for (int k = 0; k < 10; ++k)
            out[t * 10 + k] = v[k] - lse;
    }
}

extern "C" void kernel_launch(void* const* d_in, const int* in_sizes, int n_in,
                              void* d_out, int out_size, void* d_ws, size_t ws_size,
                              hipStream_t stream) {
    const float* x  = (const float*)d_in[0];   // (4,4,16,16)
    const float* qp = (const float*)d_in[1];   // (5,)
    const float* hw = (const float*)d_in[2];   // (10,1024)
    const float* hb = (const float*)d_in[3];   // (10,)
    float* out = (float*)d_out;                // (4,10)

    quanv_head_kernel<<<1, 256, 0, stream>>>(x, qp, hw, hb, out);
}